// GCNEncoder_2396591751509
// MI455X (gfx1250) — compile-verified
//
#include <hip/hip_runtime.h>
#include <hip/hip_bf16.h>

typedef float v2f __attribute__((ext_vector_type(2)));
typedef float v8f __attribute__((ext_vector_type(8)));

// ---------------------------------------------------------------------------
// Fill a float buffer with a constant (used for deg=1.0, agg=0, out=0).
// ---------------------------------------------------------------------------
__global__ void fill_kernel(float* __restrict__ p, long long n, float v) {
    long long i = (long long)blockIdx.x * blockDim.x + threadIdx.x;
    if (i < n) p[i] = v;
}

// ---------------------------------------------------------------------------
// deg[col[e]] += 1 for every edge (deg pre-initialized to 1.0 == self loop).
// ---------------------------------------------------------------------------
__global__ void degree_kernel(const long long* __restrict__ ei, int E,
                              float* __restrict__ deg) {
    int e = blockIdx.x * blockDim.x + threadIdx.x;
    if (e < E) {
        int c = (int)ei[(long long)E + e];   // col = targets
        atomicAdd(&deg[c], 1.0f);
    }
}

// dinv = rsqrt(deg), in place.
__global__ void rsqrt_kernel(float* __restrict__ d, int N) {
    int i = blockIdx.x * blockDim.x + threadIdx.x;
    if (i < N) d[i] = rsqrtf(d[i]);
}

// ---------------------------------------------------------------------------
// h = x @ W  via V_WMMA_F32_16X16X4_F32.
// One wave computes a 16x64 tile of h: 4 accumulators (16x16) x 16 K-steps.
// A-frag (16x4 f32, 2 VGPR): lanes 0-15 -> M=lane, K = {k0, k0+1};
//                            lanes 16-31 -> M=lane-16, K = {k0+2, k0+3}.
// B-frag (4x16 f32, 2 VGPR): lanes 0-15 -> N=lane, K = {k0, k0+1};
//                            lanes 16-31 -> N=lane-16, K = {k0+2, k0+3}.
// C/D (16x16 f32, 8 VGPR): VGPR j: lanes 0-15 -> (M=j, N=lane),
//                                  lanes 16-31 -> (M=j+8, N=lane-16).
// Row guard is wave-uniform -> EXEC all ones inside active waves (WMMA req).
// ---------------------------------------------------------------------------
__global__ __launch_bounds__(256) void gemm_xW_wmma(
        const float* __restrict__ x, const float* __restrict__ W,
        float* __restrict__ h, int N) {
    const int lane = threadIdx.x & 31;
    const int wave = (blockIdx.x * (blockDim.x >> 5)) + (threadIdx.x >> 5);
    const int row0 = wave * 16;
    if (row0 >= N) return;                 // wave-uniform

    const int half = lane >> 4;            // 0: lanes 0-15, 1: lanes 16-31
    const int l15  = lane & 15;

    v8f acc0 = {}, acc1 = {}, acc2 = {}, acc3 = {};
    const float* __restrict__ xrow = x + (size_t)(row0 + l15) * 64;

    #pragma unroll 4
    for (int k0 = 0; k0 < 64; k0 += 4) {
        v2f a;
        a.x = xrow[k0 + 2 * half + 0];
        a.y = xrow[k0 + 2 * half + 1];

        const float* __restrict__ Wk = W + (size_t)(k0 + 2 * half) * 64 + l15;
        v2f b0, b1, b2, b3;
        b0.x = Wk[0];        b0.y = Wk[64 + 0];
        b1.x = Wk[16];       b1.y = Wk[64 + 16];
        b2.x = Wk[32];       b2.y = Wk[64 + 32];
        b3.x = Wk[48];       b3.y = Wk[64 + 48];

        acc0 = __builtin_amdgcn_wmma_f32_16x16x4_f32(false, a, false, b0,
                                                     (short)0, acc0, false, false);
        acc1 = __builtin_amdgcn_wmma_f32_16x16x4_f32(false, a, false, b1,
                                                     (short)0, acc1, false, false);
        acc2 = __builtin_amdgcn_wmma_f32_16x16x4_f32(false, a, false, b2,
                                                     (short)0, acc2, false, false);
        acc3 = __builtin_amdgcn_wmma_f32_16x16x4_f32(false, a, false, b3,
                                                     (short)0, acc3, false, false);
    }

    float* __restrict__ hout = h + (size_t)row0 * 64;
    #pragma unroll
    for (int j = 0; j < 8; ++j) {
        const int m = j + 8 * half;
        hout[(size_t)m * 64 +  0 + l15] = acc0[j];
        hout[(size_t)m * 64 + 16 + l15] = acc1[j];
        hout[(size_t)m * 64 + 32 + l15] = acc2[j];
        hout[(size_t)m * 64 + 48 + l15] = acc3[j];
    }
}

// ---------------------------------------------------------------------------
// agg[col] += dinv[row]*dinv[col] * h[row]   (per feature dim).
// tid = e*64 + d: 64 consecutive threads share an edge -> coalesced h-row
// gather and coalesced agg-row atomics (agg fits in the 192MB L2).
// ---------------------------------------------------------------------------
__global__ void edge_scatter_kernel(const long long* __restrict__ ei, int E,
                                    const float* __restrict__ dinv,
                                    const float* __restrict__ h,
                                    float* __restrict__ agg) {
    long long t = (long long)blockIdx.x * blockDim.x + threadIdx.x;
    long long total = (long long)E * 64;
    if (t >= total) return;
    int e = (int)(t >> 6);
    int d = (int)(t & 63);
    int r = (int)ei[e];                    // row = sources
    int c = (int)ei[(long long)E + e];     // col = targets
    float w = dinv[r] * dinv[c];
    atomicAdd(&agg[(size_t)c * 64 + d], w * h[(size_t)r * 64 + d]);
}

// ---------------------------------------------------------------------------
// act = relu(agg + dinv[n]^2 * h[n] + b); out[batch[n]] += act.
// ---------------------------------------------------------------------------
__global__ void finalize_kernel(const float* __restrict__ agg,
                                const float* __restrict__ h,
                                const float* __restrict__ dinv,
                                const float* __restrict__ b,
                                const long long* __restrict__ batch,
                                float* __restrict__ out, int N) {
    long long t = (long long)blockIdx.x * blockDim.x + threadIdx.x;
    long long total = (long long)N * 64;
    if (t >= total) return;
    int n = (int)(t >> 6);
    int d = (int)(t & 63);
    float sl = dinv[n] * dinv[n];          // self-loop weight
    float v = agg[t] + sl * h[t] + b[d];
    v = fmaxf(v, 0.0f);
    int g = (int)batch[n];
    atomicAdd(&out[(size_t)g * 64 + d], v);
}

// ---------------------------------------------------------------------------
extern "C" void kernel_launch(void* const* d_in, const int* in_sizes, int n_in,
                              void* d_out, int out_size, void* d_ws, size_t ws_size,
                              hipStream_t stream) {
    const float*      x     = (const float*)d_in[0];      // [N,64] f32
    const long long*  ei    = (const long long*)d_in[1];  // [2,E]  i64
    const long long*  batch = (const long long*)d_in[2];  // [N]    i64 (sorted)
    const float*      W     = (const float*)d_in[3];      // [64,64] f32
    const float*      bias  = (const float*)d_in[4];      // [64]   f32
    float*            out   = (float*)d_out;              // [256,64] f32

    const int N = in_sizes[2];
    const int E = in_sizes[1] / 2;
    const long long ND = (long long)N * 64;

    // Workspace layout (all f32): dinv[N] | h[N*64] | agg[N*64]
    char* ws = (char*)d_ws;
    float* dinv = (float*)ws;
    float* h    = (float*)(ws + (size_t)N * sizeof(float));         // 16B-aligned (N*4 % 16 == 0)
    float* agg  = h + ND;

    const int T = 256;
    auto blocks = [](long long n, int t) { return (unsigned)((n + t - 1) / t); };

    // 1) init: deg=1 (self loop), agg=0, out=0
    fill_kernel<<<blocks(N, T), T, 0, stream>>>(dinv, N, 1.0f);
    fill_kernel<<<blocks(ND, T), T, 0, stream>>>(agg, ND, 0.0f);
    fill_kernel<<<blocks(out_size, T), T, 0, stream>>>(out, out_size, 0.0f);

    // 2) degree + rsqrt
    degree_kernel<<<blocks(E, T), T, 0, stream>>>(ei, E, dinv);
    rsqrt_kernel<<<blocks(N, T), T, 0, stream>>>(dinv, N);

    // 3) h = x @ W  (WMMA f32 16x16x4), one wave per 16 rows, 8 waves/block
    {
        int waves = (N + 15) / 16;
        unsigned g = (unsigned)((waves + 7) / 8);
        gemm_xW_wmma<<<g, 256, 0, stream>>>(x, W, h, N);
    }

    // 4) edge scatter (dominant cost, ~640MB of L2/HBM traffic)
    edge_scatter_kernel<<<blocks((long long)E * 64, T), T, 0, stream>>>(ei, E, dinv, h, agg);

    // 5) self-loop + bias + relu + pool
    finalize_kernel<<<blocks(ND, T), T, 0, stream>>>(agg, h, dinv, bias, batch, out, N);
}